// GraphSAGE_1520418422795
// MI455X (gfx1250) — compile-verified
//
#include <hip/hip_runtime.h>

typedef __attribute__((ext_vector_type(2))) float v2f;
typedef __attribute__((ext_vector_type(8))) float v8f;

#define N_NODES 50000

// ---------------------------------------------------------------------------
// Scatter-accumulate: agg[dst] += xin[src] (per feature), cnt[dst] += 1.
// One thread per (edge, feature); D = 1<<logD (128 or 32). The edge id is
// wave-uniform (feature dim spans whole waves), so route it through
// readfirstlane -> scalar loads of src/dst (s_load) instead of 32 redundant
// per-lane vector loads. Atomics are L2-resident RMWs (accumulators << 192MB).
// ---------------------------------------------------------------------------
__global__ void __launch_bounds__(256)
scatter_accum_kernel(const float* __restrict__ xin,
                     const int*   __restrict__ src,
                     const int*   __restrict__ dst,
                     float*       __restrict__ agg,
                     float*       __restrict__ cnt,
                     int E, int logD)
{
    const long long tid = (long long)blockIdx.x * 256 + threadIdx.x;
    const long long total = (long long)E << logD;
    if (tid >= total) return;
    const int D = 1 << logD;
    const int d = (int)(tid & (D - 1));
    // wave-uniform edge index -> scalar register -> scalar memory path
    const int e = __builtin_amdgcn_readfirstlane((int)(tid >> logD));
    const int s = src[e];
    const int t = dst[e];
    atomicAdd(agg + (size_t)t * D + d, xin[(size_t)s * D + d]);
    if (cnt != nullptr && d == 0) atomicAdd(cnt + t, 1.0f);
}

// ---------------------------------------------------------------------------
// out[row, 0:32] = maybe_relu( (agg[row,:]/max(cnt[row],1)) @ Wl + bias
//                              + xin[row,:] @ Wr )
// One wave32 computes a 16x32 output strip via V_WMMA_F32_16X16X4_F32.
// N_NODES = 3125 * 16 exactly -> every WMMA wave has EXEC all-1s.
//
// Fragment layouts (ISA 7.12.2, fp32 16x16x4):
//   A (16x4, MxK): lanes 0-15 -> M=lane, v0=K0 v1=K1 ; lanes 16-31 -> K2,K3
//   B (4x16, KxN): lanes 0-15 -> N=lane, v0=K0 v1=K1 ; lanes 16-31 -> K2,K3
//   C/D (16x16):   lanes 0-15 -> N=lane, vgpr r -> M=r ; lanes 16-31 -> M=8+r
// ---------------------------------------------------------------------------
template <int K, bool RELU>
__global__ void __launch_bounds__(128)
sage_linear_wmma(const float* __restrict__ agg,
                 const float* __restrict__ cnt,
                 const float* __restrict__ xin,
                 const float* __restrict__ Wl,    // [K x 32] row-major
                 const float* __restrict__ bias,  // [32]
                 const float* __restrict__ Wr,    // [K x 32] row-major
                 float*       __restrict__ out,   // [N x 32]
                 int nTiles)
{
    const int lane = threadIdx.x & 31;
    const int wave = threadIdx.x >> 5;
    const int tile = blockIdx.x * 4 + wave;   // wave-uniform guard: EXEC stays all-1s
    if (tile >= nTiles) return;

    const int m    = lane & 15;   // M for A-frag / N for B- and C-frags
    const int half = lane >> 4;   // K sub-pair selector
    const int row  = tile * 16 + m;

    // 1/max(cnt,1): v_rcp_f32 + one Newton step (~1 ulp) instead of the
    // full IEEE div_scale/div_fmas/div_fixup sequence.
    const float c    = fmaxf(cnt[row], 1.0f);
    float       invc = __builtin_amdgcn_rcpf(c);
    invc = invc * (2.0f - c * invc);

    v8f acc0 = {};   // columns 0..15
    v8f acc1 = {};   // columns 16..31

    const float* arow = agg + (size_t)row * K;
    const float* xrow = xin + (size_t)row * K;

    // (agg/cnt) @ Wl  — fold the mean into the A fragment load
#pragma unroll
    for (int k0 = 0; k0 < K; k0 += 4) {
        const int ka = k0 + half * 2;
        v2f a;  a.x = arow[ka] * invc;          a.y = arow[ka + 1] * invc;
        v2f b0; b0.x = Wl[ka * 32 + m];         b0.y = Wl[(ka + 1) * 32 + m];
        v2f b1; b1.x = Wl[ka * 32 + 16 + m];    b1.y = Wl[(ka + 1) * 32 + 16 + m];
        acc0 = __builtin_amdgcn_wmma_f32_16x16x4_f32(false, a, false, b0, (short)0, acc0, false, false);
        acc1 = __builtin_amdgcn_wmma_f32_16x16x4_f32(false, a, false, b1, (short)0, acc1, false, false);
    }

    // x @ Wr
#pragma unroll
    for (int k0 = 0; k0 < K; k0 += 4) {
        const int ka = k0 + half * 2;
        v2f a;  a.x = xrow[ka];                 a.y = xrow[ka + 1];
        v2f b0; b0.x = Wr[ka * 32 + m];         b0.y = Wr[(ka + 1) * 32 + m];
        v2f b1; b1.x = Wr[ka * 32 + 16 + m];    b1.y = Wr[(ka + 1) * 32 + 16 + m];
        acc0 = __builtin_amdgcn_wmma_f32_16x16x4_f32(false, a, false, b0, (short)0, acc0, false, false);
        acc1 = __builtin_amdgcn_wmma_f32_16x16x4_f32(false, a, false, b1, (short)0, acc1, false, false);
    }

    const float bv0 = bias[m];
    const float bv1 = bias[16 + m];

    float* orow = out + (size_t)tile * 16 * 32;
#pragma unroll
    for (int r = 0; r < 8; ++r) {
        const int M = r + 8 * half;
        float v0 = acc0[r] + bv0;
        float v1 = acc1[r] + bv1;
        if (RELU) { v0 = fmaxf(v0, 0.0f); v1 = fmaxf(v1, 0.0f); }
        orow[M * 32 + m]      = v0;
        orow[M * 32 + 16 + m] = v1;
    }
}

// ---------------------------------------------------------------------------
// Orchestration. Workspace layout (floats):
//   agg1 [N*128] | cnt [N] | h [N*32] | agg2 [N*32]   (~38.6 MB)
// ---------------------------------------------------------------------------
extern "C" void kernel_launch(void* const* d_in, const int* in_sizes, int n_in,
                              void* d_out, int out_size, void* d_ws, size_t ws_size,
                              hipStream_t stream)
{
    const float* x   = (const float*)d_in[0];
    const int*   ei  = (const int*)d_in[1];   // [2, E] (JAX default x64-off -> int32)
    const float* W1l = (const float*)d_in[2];
    const float* b1  = (const float*)d_in[3];
    const float* W1r = (const float*)d_in[4];
    const float* W2l = (const float*)d_in[5];
    const float* b2  = (const float*)d_in[6];
    const float* W2r = (const float*)d_in[7];
    float* out = (float*)d_out;

    const int N  = N_NODES;
    const int E  = in_sizes[1] / 2;
    const int D1 = in_sizes[0] / N;   // 128
    const int D2 = 32;

    float* agg1 = (float*)d_ws;
    float* cnt  = agg1 + (size_t)N * D1;
    float* h    = cnt  + N;
    float* agg2 = h    + (size_t)N * D2;

    // zero accumulators every call (graph-capture-safe memset nodes)
    hipMemsetAsync(agg1, 0, sizeof(float) * ((size_t)N * D1 + N), stream);
    hipMemsetAsync(agg2, 0, sizeof(float) * (size_t)N * D2, stream);

    const int* src = ei;
    const int* dst = ei + E;

    {   // layer-1 neighbor sum + degree count
        const long long total = (long long)E * D1;
        const int blocks = (int)((total + 255) / 256);
        scatter_accum_kernel<<<blocks, 256, 0, stream>>>(x, src, dst, agg1, cnt, E, 7);
    }
    {   // layer-1 linear (+bias on Wl path) + ReLU
        const int nTiles = N / 16;                 // 3125
        const int blocks = (nTiles + 3) / 4;
        sage_linear_wmma<128, true><<<blocks, 128, 0, stream>>>(
            agg1, cnt, x, W1l, b1, W1r, h, nTiles);
    }
    {   // layer-2 neighbor sum (same degrees)
        const long long total = (long long)E * D2;
        const int blocks = (int)((total + 255) / 256);
        scatter_accum_kernel<<<blocks, 256, 0, stream>>>(h, src, dst, agg2, nullptr, E, 5);
    }
    {   // layer-2 linear, no ReLU -> final output
        const int nTiles = N / 16;
        const int blocks = (nTiles + 3) / 4;
        sage_linear_wmma<32, false><<<blocks, 128, 0, stream>>>(
            agg2, cnt, h, W2l, b2, W2r, out, nTiles);
    }
}